// PSEModule_24249385353824
// MI455X (gfx1250) — compile-verified
//
#include <hip/hip_runtime.h>

// Fused squeeze-excite kernel for MI455X (gfx1250, wave32).
// One pass over x: NT-load tile -> channel mean -> WMMA fp32 16x16x4 GEMM
// for the 16->256 excitation -> sigmoid -> multiply -> NT-store.

typedef float v2f __attribute__((ext_vector_type(2)));
typedef float v4f __attribute__((ext_vector_type(4)));
typedef float v8f __attribute__((ext_vector_type(8)));

#define TPB 256          // 8 wave32 waves
#define NPIX 32          // pixels per block tile
#define C_CH 256         // channels
#define HID 16           // hidden (K of the GEMM)
#define HW 4096          // H*W per (b,c) plane
#define SC_STRIDE 36     // padded LDS row stride for scale (dwords); 16B-aligned, bank-disjoint halves
#define W2T_STRIDE 257   // padded LDS row stride for transposed w2 (conflict-free)

__global__ __launch_bounds__(TPB) void se_fused_wmma(
    const float* __restrict__ x, const float* __restrict__ w1,
    const float* __restrict__ b1, const float* __restrict__ w2,
    const float* __restrict__ b2, float* __restrict__ out)
{
    __shared__ float w2t[HID * W2T_STRIDE];          // w2 transposed: [k][c]
    __shared__ float b2s[C_CH];
    __shared__ float w1s[HID], b1s[HID];
    __shared__ float s_lds[NPIX];                    // per-pixel channel mean
    __shared__ v4f   part[TPB];                      // partial channel sums
    __shared__ float scale_lds[C_CH * SC_STRIDE];    // sigmoid gate, [c][p]

    const int tid  = threadIdx.x;
    const int lane = tid & 31;
    const int wv   = tid >> 5;      // wave id 0..7
    const int half = lane >> 4;     // 0: lanes 0-15, 1: lanes 16-31
    const int li   = lane & 15;

    // ---- tile base: 32 contiguous pixels of one batch image ----
    const long pixBase = (long)blockIdx.x * NPIX;    // flat over B*H*W
    const int  b   = (int)(pixBase >> 12);           // / 4096
    const int  hw0 = (int)(pixBase & 4095);
    const float* xb = x   + ((long)b * C_CH) * HW + hw0;
    float*       ob = out + ((long)b * C_CH) * HW + hw0;

    // ---- preload small tensors to LDS ----
    if (tid < HID) { w1s[tid] = w1[tid]; b1s[tid] = b1[tid]; }
    b2s[tid] = b2[tid];
    #pragma unroll
    for (int i = 0; i < 16; ++i) {                   // coalesced read of w2 (256x16)
        int idx = i * 256 + tid;                     // flat = c*16 + k
        int c = idx >> 4, k = idx & 15;
        w2t[k * W2T_STRIDE + c] = w2[idx];
    }

    // ---- load x tile (256 ch x 32 pix) into registers; NT so it streams ----
    const int cBase = tid >> 3;                      // 0..31
    const int p0    = (tid & 7) * 4;                 // pixel group 0,4,...,28
    v4f xv[8];
    v4f psum = {0.f, 0.f, 0.f, 0.f};
    #pragma unroll
    for (int i = 0; i < 8; ++i) {
        int c = cBase + 32 * i;                      // each c covered exactly once
        xv[i] = __builtin_nontemporal_load((const v4f*)(xb + (long)c * HW + p0));
        psum += xv[i];
    }
    part[tid] = psum;
    __syncthreads();

    // ---- finish channel mean: s_lds[p] = (1/256) * sum_c x[c][p] ----
    if (tid < NPIX) {
        const float* praw = (const float*)part;      // praw[u*32 + p] covers pixel p
        float ssum = 0.f;
        #pragma unroll
        for (int u = 0; u < 32; ++u) ssum += praw[u * 32 + tid];
        s_lds[tid] = ssum * (1.0f / 256.0f);
    }
    __syncthreads();

    // ---- GEMM: D(256x32) = w2(256x16) x h(16x32) via V_WMMA_F32_16X16X4_F32 ----
    // A layout (16x4 f32): lane%16 = M, K = 2*(lane/16) + vgpr  (K base += 4 per step)
    // B layout (4x16 f32): lane%16 = N, same (lane-half, vgpr) -> K mapping
    v2f Bop[2][4];                                   // [n-tile][k-chunk]
    #pragma unroll
    for (int n = 0; n < 2; ++n) {
        float sv = s_lds[n * 16 + li];
        #pragma unroll
        for (int q = 0; q < 4; ++q) {
            int k0 = 4 * q + 2 * half;
            Bop[n][q].x = fmaxf(sv * w1s[k0]     + b1s[k0],     0.f);  // h[k0][p]
            Bop[n][q].y = fmaxf(sv * w1s[k0 + 1] + b1s[k0 + 1], 0.f);  // h[k0+1][p]
        }
    }
    v2f Aop[2][4];                                   // [m-sub][k-chunk]; wave owns M-tiles 2w,2w+1
    #pragma unroll
    for (int im = 0; im < 2; ++im) {
        int c = (2 * wv + im) * 16 + li;
        #pragma unroll
        for (int q = 0; q < 4; ++q) {
            int k0 = 4 * q + 2 * half;
            Aop[im][q].x = w2t[k0 * W2T_STRIDE + c];
            Aop[im][q].y = w2t[(k0 + 1) * W2T_STRIDE + c];
        }
    }

    v8f acc[2][2] = {};                              // [m-sub][n-tile]
    #pragma unroll
    for (int im = 0; im < 2; ++im)
        #pragma unroll
        for (int n = 0; n < 2; ++n)
            #pragma unroll
            for (int q = 0; q < 4; ++q)
                acc[im][n] = __builtin_amdgcn_wmma_f32_16x16x4_f32(
                    /*neg_a=*/false, Aop[im][q],
                    /*neg_b=*/false, Bop[n][q],
                    /*c_mod=*/(short)0, acc[im][n],
                    /*reuse_a=*/false, /*reuse_b=*/false);

    // ---- sigmoid(D + b2) -> scale_lds;  D vgpr r: M = r + 8*half, N = lane%16 ----
    #pragma unroll
    for (int im = 0; im < 2; ++im) {
        int mbase = (2 * wv + im) * 16 + 8 * half;
        #pragma unroll
        for (int n = 0; n < 2; ++n) {
            int p = n * 16 + li;
            #pragma unroll
            for (int r = 0; r < 8; ++r) {
                int c = mbase + r;
                float v = acc[im][n][r] + b2s[c];
                scale_lds[c * SC_STRIDE + p] = 1.0f / (1.0f + __expf(-v));
            }
        }
    }
    __syncthreads();

    // ---- apply gate to register-resident x tile; stream out NT ----
    #pragma unroll
    for (int i = 0; i < 8; ++i) {
        int c = cBase + 32 * i;
        v4f sc = *(const v4f*)&scale_lds[c * SC_STRIDE + p0];
        __builtin_nontemporal_store(xv[i] * sc, (v4f*)(ob + (long)c * HW + p0));
    }
}

extern "C" void kernel_launch(void* const* d_in, const int* in_sizes, int n_in,
                              void* d_out, int out_size, void* d_ws, size_t ws_size,
                              hipStream_t stream) {
    const float* x  = (const float*)d_in[0];
    const float* w1 = (const float*)d_in[1];
    const float* b1 = (const float*)d_in[2];
    const float* w2 = (const float*)d_in[3];
    const float* b2 = (const float*)d_in[4];
    float* out = (float*)d_out;

    const int npix_total = in_sizes[0] / C_CH;       // B*H*W = 131072
    const int blocks = npix_total / NPIX;            // 4096
    se_fused_wmma<<<blocks, TPB, 0, stream>>>(x, w1, b1, w2, b2, out);
}